// FairMamba_35347580846761
// MI455X (gfx1250) — compile-verified
//
#include <hip/hip_runtime.h>

typedef __attribute__((ext_vector_type(2))) float v2f;
typedef __attribute__((ext_vector_type(8))) float v8f;

static constexpr int T_SEQ = 32, IN_DIM = 64, HID = 128, OUT_DIM = 64, STEPS = 20;
static constexpr float LOG2E = 1.4426950408889634f;

// One wave handles a 16-batch tile. h is kept in registers in WMMA C/D layout:
// h[tile t][vgpr r] at lane L = h[batch r + 8*(L>=16)][hidden 16*t + (L&15)].
__global__ __launch_bounds__(256)
void fairmamba_wmma_kernel(const float* __restrict__ x,
                           const float* __restrict__ W_in,
                           const float* __restrict__ b_in,
                           const float* __restrict__ Avec,
                           const float* __restrict__ Bssm,
                           const float* __restrict__ Cvec,
                           const float* __restrict__ Dvec,
                           const float* __restrict__ W_dt,
                           const float* __restrict__ b_dt,
                           const float* __restrict__ W_dec,
                           const float* __restrict__ b_dec,
                           float* __restrict__ out)
{
    __shared__ __align__(16) float wdec_s[OUT_DIM * 130];   // padded rows: bank-friendly
    __shared__ __align__(16) float tbuf[8][16 * 18];        // per-wave transpose buffer

    const int tid  = threadIdx.x;
    const int wave = tid >> 5;
    const int lane = tid & 31;
    const int n0   = lane & 15;   // column index inside a 16-wide tile
    const int hi   = lane >> 4;   // half-wave select

    // Stage W_dec [64 x 128] into padded LDS (coalesced).
    for (int idx = tid; idx < OUT_DIM * HID; idx += 256) {
        wdec_s[(idx >> 7) * 130 + (idx & 127)] = W_dec[idx];
    }
    __syncthreads();

    const int batch_base = blockIdx.x * 128 + wave * 16;

    // Per-lane SSM constants over the 8 hidden tiles (hidden = 16*t + n0).
    float a2[8], binv[8], cc[8], dd[8], wdt[8];
#pragma unroll
    for (int t = 0; t < 8; ++t) {
        const int n = t * 16 + n0;
        const float av = Avec[n];
        a2[t]   = av * LOG2E;                 // fold log2(e) for v_exp_f32
        binv[t] = Bssm[n] / (av + 1e-6f);
        cc[t]   = Cvec[n];
        dd[t]   = Dvec[n];
        wdt[t]  = W_dt[n];
    }
    float bdec[4];
#pragma unroll
    for (int j = 0; j < 4; ++j) bdec[j] = b_dec[j * 16 + n0];
    const float bdt = b_dt[0];

    // ---- GEMM1: h0 = x_last[16x64] * W_in^T[64x128] via V_WMMA_F32_16X16X4_F32.
    // A fragment (16x4): lane m = n0 (batch), K = 4t + 2*hi + {0,1}  -> one b64 load.
    v2f ax[16];
    {
        const float* xrow = x + (size_t)(batch_base + n0) * (T_SEQ * IN_DIM)
                              + (size_t)(T_SEQ - 1) * IN_DIM;
        const v2f* xr2 = (const v2f*)xrow;
#pragma unroll
        for (int t = 0; t < 16; ++t) ax[t] = xr2[2 * t + hi];
    }

    v8f h[8];
#pragma unroll
    for (int j = 0; j < 8; ++j) {
        v8f acc = {};
        // B fragment (4x16): lane n = 16j+n0 (hidden), K = 4t + 2*hi + {0,1}.
        const v2f* wr2 = (const v2f*)(W_in + (size_t)(j * 16 + n0) * IN_DIM);
#pragma unroll
        for (int t = 0; t < 16; ++t) {
            acc = __builtin_amdgcn_wmma_f32_16x16x4_f32(false, ax[t], false, wr2[2 * t + hi],
                                                        (short)0, acc, false, false);
        }
        const float bin = b_in[j * 16 + n0];
#pragma unroll
        for (int r = 0; r < 8; ++r) acc[r] += bin;
        h[j] = acc;
    }

    float* tb = tbuf[wave];

#pragma unroll 1
    for (int s = 0; s < STEPS; ++s) {
        // dt[b] = exp(sum_n h[b][n]*W_dt[n] + b_dt): per-lane FMAs + half-wave xor reduce.
        float dtb[8];
#pragma unroll
        for (int r = 0; r < 8; ++r) {
            float p = 0.f;
#pragma unroll
            for (int t = 0; t < 8; ++t) p = __builtin_fmaf(h[t][r], wdt[t], p);
            p += __shfl_xor(p, 1, 32);
            p += __shfl_xor(p, 2, 32);
            p += __shfl_xor(p, 4, 32);
            p += __shfl_xor(p, 8, 32);               // masks < 16: halves stay separate
            dtb[r] = __builtin_exp2f((p + bdt) * LOG2E);
        }

        // Elementwise SSM update; build transposed A fragments of y through LDS.
        v2f af[32];
#pragma unroll
        for (int t = 0; t < 8; ++t) {
#pragma unroll
            for (int r = 0; r < 8; ++r) {
                const float hv = h[t][r];
                const float dA = __builtin_exp2f(dtb[r] * a2[t]);
                const float sc = __builtin_fmaf(dA - 1.f, binv[t], dA); // dA + dB/B-fold
                const float hn = sc * hv;                                // h_new
                h[t][r] = hn;
                // y = h_new*C + h*D, stored [batch][hidden-in-tile], stride 18
                tb[(r + 8 * hi) * 18 + n0] = __builtin_fmaf(hn, cc[t], hv * dd[t]);
            }
            // Reload as A fragments: lane m = n0 (batch), K = 4q + 2*hi + {0,1}.
            const v2f* tr2 = (const v2f*)(tb + n0 * 18);
#pragma unroll
            for (int q = 0; q < 4; ++q) af[t * 4 + q] = tr2[2 * q + hi];
        }

        // pred = y[16x128] * W_dec^T[128x64]: 4 N-tiles x 32 K-slices of WMMA f32.
        const size_t obase = ((size_t)batch_base * STEPS + s) * OUT_DIM + n0;
#pragma unroll
        for (int j = 0; j < 4; ++j) {
            v8f acc = {};
            const v2f* wr2 = (const v2f*)(wdec_s + (j * 16 + n0) * 130);
#pragma unroll
            for (int t = 0; t < 32; ++t) {
                acc = __builtin_amdgcn_wmma_f32_16x16x4_f32(false, af[t], false, wr2[2 * t + hi],
                                                            (short)0, acc, false, false);
            }
#pragma unroll
            for (int r = 0; r < 8; ++r) {
                out[obase + (size_t)(r + 8 * hi) * (STEPS * OUT_DIM) + j * 16] = acc[r] + bdec[j];
            }
        }
    }
}

extern "C" void kernel_launch(void* const* d_in, const int* in_sizes, int n_in,
                              void* d_out, int out_size, void* d_ws, size_t ws_size,
                              hipStream_t stream) {
    const float* x     = (const float*)d_in[0];
    const float* W_in  = (const float*)d_in[1];
    const float* b_in  = (const float*)d_in[2];
    const float* A     = (const float*)d_in[3];
    const float* Bssm  = (const float*)d_in[4];
    const float* C     = (const float*)d_in[5];
    const float* D     = (const float*)d_in[6];
    const float* W_dt  = (const float*)d_in[7];
    const float* b_dt  = (const float*)d_in[8];
    const float* W_dec = (const float*)d_in[9];
    const float* b_dec = (const float*)d_in[10];
    float* out = (float*)d_out;

    const int B = in_sizes[0] / (T_SEQ * IN_DIM);   // 65536
    const int blocks = B / 128;                     // 16 batches/wave * 8 waves
    fairmamba_wmma_kernel<<<blocks, 256, 0, stream>>>(x, W_in, b_in, A, Bssm, C, D,
                                                      W_dt, b_dt, W_dec, b_dec, out);
}